// PatchMoments_6932077215795
// MI455X (gfx1250) — compile-verified
//
#include <hip/hip_runtime.h>

typedef float v2f __attribute__((ext_vector_type(2)));
typedef float v8f __attribute__((ext_vector_type(8)));

#define IMG_H 512
#define IMG_W 512
#define NBATCH 16
#define TS 16
#define KSZ 5
#define PADR 2
#define HALO (TS + KSZ - 1) /* 20 */

__device__ __forceinline__ int refl(int i) {
    i = (i < 0) ? -i : i;
    i = (i >= IMG_H) ? (2 * IMG_H - 2 - i) : i;
    return i;
}

__global__ __launch_bounds__(256)
void patch_moments_wmma(const float* __restrict__ x, float* __restrict__ out) {
    // Per-wave LDS staging for the vertical-pass result T (16 rows x 20 cols, 2 moments).
    // Row stride 20 floats: conflict-free for both D-layout stores and A-layout loads.
    __shared__ float Tlds[8][2][16][HALO];

    const int lane  = threadIdx.x & 31;
    const int wave  = threadIdx.x >> 5;
    const int tile  = blockIdx.x * 8 + wave;

    const int b  = tile >> 10;        // 1024 tiles per image (32x32)
    const int tr = (tile >> 5) & 31;
    const int tc = tile & 31;
    const int r0 = tr * TS;
    const int c0 = tc * TS;

    const float* __restrict__ xb = x + (size_t)b * (IMG_H * IMG_W);

    const int n     = lane & 15;   // B/D column within tile, also A row (m)
    const int khalf = lane >> 4;   // which half of the K striping / M striping
    const int m     = n;

    // ---------------- Pass 1: vertical band matmul  T = V(16x20) * Xp(20x20) ----------------
    v8f dsum0 = {}, dsum1 = {};    // col-groups 0..15 and 16..19(+junk)
    v8f dsq0  = {}, dsq1  = {};

    #pragma unroll
    for (int g = 0; g < 2; ++g) {
        const int cpad = g * 16 + n;                 // padded col 0..31 (>=20 -> junk, discarded)
        const int gcol = refl(c0 - PADR + cpad);     // always a valid in-bounds column
        #pragma unroll
        for (int kc = 0; kc < 5; ++kc) {
            v2f a, bx, bxx;
            #pragma unroll
            for (int p = 0; p < 2; ++p) {
                const int k = 4 * kc + 2 * khalf + p;        // padded row 0..19
                // A = V band slice: V[i][k] = 1 iff i <= k <= i+4
                a[p] = (k >= m && k <= m + 4) ? 1.0f : 0.0f;
                // B = data slice (reflect-padded), coalesced 64B per half-wave
                const int grow = refl(r0 - PADR + k);
                const float v  = xb[grow * IMG_W + gcol];
                bx[p]  = v;
                bxx[p] = v * v;
            }
            if (g == 0) {
                dsum0 = __builtin_amdgcn_wmma_f32_16x16x4_f32(false, a, false, bx,  (short)0, dsum0, false, false);
                dsq0  = __builtin_amdgcn_wmma_f32_16x16x4_f32(false, a, false, bxx, (short)0, dsq0,  false, false);
            } else {
                dsum1 = __builtin_amdgcn_wmma_f32_16x16x4_f32(false, a, false, bx,  (short)0, dsum1, false, false);
                dsq1  = __builtin_amdgcn_wmma_f32_16x16x4_f32(false, a, false, bxx, (short)0, dsq1,  false, false);
            }
        }
    }

    // Store T to LDS (D layout: element (M = v + 8*khalf, N = n) in vgpr v)
    #pragma unroll
    for (int vv = 0; vv < 8; ++vv) {
        const int row = vv + 8 * khalf;
        Tlds[wave][0][row][n] = dsum0[vv];
        Tlds[wave][1][row][n] = dsq0[vv];
        if (n < HALO - 16) {                 // cols 16..19 from group 1
            Tlds[wave][0][row][16 + n] = dsum1[vv];
            Tlds[wave][1][row][16 + n] = dsq1[vv];
        }
    }
    // Cross-lane RAW within the wave: drain DS before reading other lanes' T values.
    asm volatile("s_wait_dscnt 0" ::: "memory");

    // ---------------- Pass 2: horizontal band matmul  Y = T(16x20) * H(20x16) ----------------
    v8f s1 = {}, s2 = {};
    #pragma unroll
    for (int kc = 0; kc < 5; ++kc) {
        v2f asum, asq, bb;
        #pragma unroll
        for (int p = 0; p < 2; ++p) {
            const int k = 4 * kc + 2 * khalf + p;            // 0..19 (< HALO always)
            asum[p] = Tlds[wave][0][m][k];
            asq[p]  = Tlds[wave][1][m][k];
            // B = H band slice: H[k][j] = 1 iff j <= k <= j+4
            bb[p] = (k >= n && k <= n + 4) ? 1.0f : 0.0f;
        }
        s1 = __builtin_amdgcn_wmma_f32_16x16x4_f32(false, asum, false, bb, (short)0, s1, false, false);
        s2 = __builtin_amdgcn_wmma_f32_16x16x4_f32(false, asq,  false, bb, (short)0, s2, false, false);
    }

    // ---------------- Finalize: mean and unbiased variance, store ----------------
    const float inv_n   = 1.0f / 25.0f;
    const float inv_nm1 = 1.0f / 24.0f;
    float* __restrict__ avg_out = out;
    float* __restrict__ var_out = out + (size_t)NBATCH * IMG_H * IMG_W;
    const size_t base = (size_t)b * (IMG_H * IMG_W);

    #pragma unroll
    for (int vv = 0; vv < 8; ++vv) {
        const int row = r0 + vv + 8 * khalf;
        const int col = c0 + n;
        const float ssum = s1[vv];
        const float a    = ssum * inv_n;
        const float var  = (s2[vv] - ssum * a) * inv_nm1;
        const size_t idx = base + (size_t)(row * IMG_W + col);
        avg_out[idx] = a;
        var_out[idx] = var;
    }
}

extern "C" void kernel_launch(void* const* d_in, const int* in_sizes, int n_in,
                              void* d_out, int out_size, void* d_ws, size_t ws_size,
                              hipStream_t stream) {
    (void)in_sizes; (void)n_in; (void)out_size; (void)d_ws; (void)ws_size;
    const float* x = (const float*)d_in[0];
    float* out = (float*)d_out;
    // 16384 tiles (16 images x 32x32 tiles), 8 waves (tiles) per 256-thread block.
    patch_moments_wmma<<<dim3(2048), dim3(256), 0, stream>>>(x, out);
}